// SlidingMode_32916629357059
// MI455X (gfx1250) — compile-verified
//
#include <hip/hip_runtime.h>
#include <math.h>

#define DELTA 1e-3f

// gfx1250 hardware tanh transcendental (TRANS op v_tanh_f32).
// Guaranteed to lower to the hardware instruction on both paths.
__device__ __forceinline__ float hw_tanhf(float x) {
#if __has_builtin(__builtin_amdgcn_tanhf)
    return __builtin_amdgcn_tanhf(x);
#else
    float r;
    asm("v_tanh_f32 %0, %1" : "=v"(r) : "v"(x));
    return r;
#endif
}

// Broadcast lane `lane`'s value of v to all lanes (v_readlane_b32).
__device__ __forceinline__ float lane_bcast(float v, int lane) {
#if __has_builtin(__builtin_amdgcn_readlane)
    return __uint_as_float(__builtin_amdgcn_readlane(__float_as_uint(v), lane));
#else
    return __shfl(v, lane, 32);
#endif
}

// ---------------------------------------------------------------------------
// Kernel 1: desired trajectory (parallel, bandwidth-trivial: ~8MB -> L2)
//   desired[i] = (sin(i*delta), cos(i*delta)), t computed as fp32 i*delta
//   also writes all_system_variables row 0 = initial_values.
// ---------------------------------------------------------------------------
__global__ void SlidingMode_desired_kernel(const float* __restrict__ iv,
                                           float* __restrict__ out, int n) {
    int i = blockIdx.x * blockDim.x + threadIdx.x;
    if (i < n) {
        float t = (float)i * DELTA;
        float s, c;
        sincosf(t, &s, &c);                 // precise (large-arg reduction)
        float* desired = out + 2 * (size_t)n;
        desired[2 * (size_t)i + 0] = s;
        desired[2 * (size_t)i + 1] = c;
    }
    if (i == 0) {
        out[0] = iv[0];
        out[1] = iv[1];
    }
}

// ---------------------------------------------------------------------------
// Kernel 2: the serial Euler scan, one wave32.
// All 32 lanes redundantly compute the (wave-uniform) dependent chain.
// Per block of 32 steps:
//   - one coalesced, double-buffered b64 load of 32 desired pairs (L2 hit,
//     latency hidden behind ~32 serial steps),
//   - v_readlane (imm lane, fully unrolled) feeds step m's desired values,
//   - (lane==m) selects stage the 32 produced states for one coalesced store.
// Critical path per step: sub -> add -> mul10 -> v_tanh_f32 -> fma -> fma.
// ---------------------------------------------------------------------------
__global__ void __launch_bounds__(32)
SlidingMode_scan_kernel(const float* __restrict__ params,
                        const float* __restrict__ iv,
                        float* __restrict__ out, int n) {
    const int lane = threadIdx.x;                  // wave32
    const float k1 = params[0], k2 = params[1], k3 = params[2];

    const float* __restrict__ desired = out + 2 * (size_t)n;
    float* __restrict__ traj = out;                // rows 1..n-1 written here

    float sv0 = iv[0];
    float sv1 = iv[1];
    float loss = 0.0f;

    const int steps = n - 1;                       // 1,000,000 for n=1,000,001
    const int nfull = steps / 32;                  // full 32-step blocks

    // preload block 0 (steps 1..32): lane l holds desired[1 + l]
    float d0 = 0.0f, d1 = 0.0f;
    {
        int idx = 1 + lane;
        if (idx < n) {
            d0 = desired[2 * (size_t)idx + 0];
            d1 = desired[2 * (size_t)idx + 1];
        }
    }

    for (int b = 0; b < nfull; ++b) {
        const int base = 1 + b * 32;

        // prefetch next block (double buffer; wait lands next iteration)
        float nd0 = 0.0f, nd1 = 0.0f;
        {
            int nidx = base + 32 + lane;
            if (nidx < n) {
                nd0 = desired[2 * (size_t)nidx + 0];
                nd1 = desired[2 * (size_t)nidx + 1];
            }
        }

        float o0 = 0.0f, o1 = 0.0f;                // per-lane output staging
#pragma unroll
        for (int m = 0; m < 32; ++m) {
            const float dd0 = lane_bcast(d0, m);   // v_readlane, imm lane
            const float dd1 = lane_bcast(d1, m);

            // exact reference op order
            const float e0 = sv0 - dd0;
            const float e1 = sv1 - dd1;
            const float T  = hw_tanhf(10.0f * (e0 + e1));   // v_tanh_f32
            const float u  = -k1 * e0 - k2 * e1 - k3 * T;

            const float nsv0 = sv0 + DELTA * sv1;  // dx1 = x2 (old sv1)
            const float nsv1 = sv1 + DELTA * u;    // dx2 = u
            sv0 = nsv0;
            sv1 = nsv1;

            const float a = sv0 - dd0;
            const float c = sv1 - dd1;
            loss += a * a + c * c;                 // fp32, reference order

            // stage step m's (uniform) state into lane m: cmp + cndmask,
            // off the critical path, co-issues with the chain.
            const bool mine = (lane == m);
            o0 = mine ? sv0 : o0;
            o1 = mine ? sv1 : o1;
        }

        // one coalesced 256B store for the 32 steps of this block
        {
            int sidx = base + lane;
            traj[2 * (size_t)sidx + 0] = o0;
            traj[2 * (size_t)sidx + 1] = o1;
        }

        d0 = nd0;
        d1 = nd1;
    }

    // tail (steps % 32 != 0; empty for n = 1,000,001)
    for (int i = 1 + nfull * 32; i < n; ++i) {
        const float dd0 = desired[2 * (size_t)i + 0];
        const float dd1 = desired[2 * (size_t)i + 1];
        const float e0 = sv0 - dd0;
        const float e1 = sv1 - dd1;
        const float T  = hw_tanhf(10.0f * (e0 + e1));
        const float u  = -k1 * e0 - k2 * e1 - k3 * T;
        const float nsv0 = sv0 + DELTA * sv1;
        const float nsv1 = sv1 + DELTA * u;
        sv0 = nsv0;
        sv1 = nsv1;
        const float a = sv0 - dd0;
        const float c = sv1 - dd1;
        loss += a * a + c * c;
        if (lane == 0) {
            traj[2 * (size_t)i + 0] = sv0;
            traj[2 * (size_t)i + 1] = sv1;
        }
    }

    if (lane == 0) {
        out[(size_t)4 * n] = loss;                 // total_loss
    }
}

extern "C" void kernel_launch(void* const* d_in, const int* in_sizes, int n_in,
                              void* d_out, int out_size, void* d_ws, size_t ws_size,
                              hipStream_t stream) {
    (void)in_sizes; (void)n_in; (void)d_ws; (void)ws_size;

    const float* params = (const float*)d_in[0];   // k1, k2, k3
    const float* iv     = (const float*)d_in[1];   // x1(0), x2(0)
    float* out          = (float*)d_out;

    // out = asv[n*2] ++ desired[n*2] ++ loss[1]  =>  n = (out_size-1)/4
    const int n = (out_size - 1) / 4;
    if (n <= 0) return;

    const int threads = 256;
    const int blocks  = (n + threads - 1) / threads;
    SlidingMode_desired_kernel<<<blocks, threads, 0, stream>>>(iv, out, n);
    SlidingMode_scan_kernel<<<1, 32, 0, stream>>>(params, iv, out, n);
}